// TimeAwareAttention_21878563406313
// MI455X (gfx1250) — compile-verified
//
#include <hip/hip_runtime.h>

// ---------------------------------------------------------------------------
// Time-aware attention, MI455X (gfx1250, wave32).
// Memory-bound: tK/tV are 256MB each -> stream once (non-temporal), fuse rest.
// WMMA (fp32 16x16x4) used for the dense projection GEMMs and base QK^T.
// ---------------------------------------------------------------------------

typedef __attribute__((ext_vector_type(2))) float v2f;
typedef __attribute__((ext_vector_type(4))) float v4f;
typedef __attribute__((ext_vector_type(8))) float v8f;

#define BB      4
#define LLEN    256
#define HH      256
#define NHEADS  4
#define HS      64
#define QT      4          // queries per workgroup in fused kernel
#define NEG_INF (-4294967295.0f)

__device__ __forceinline__ v8f wmma_f32_k4(v2f a, v2f b, v8f c) {
  // V_WMMA_F32_16X16X4_F32: D = A(16x4) * B(4x16) + C(16x16), all fp32.
  return __builtin_amdgcn_wmma_f32_16x16x4_f32(
      /*neg_a=*/false, a, /*neg_b=*/false, b,
      /*c_mod=*/(short)0, c, /*reuse_a=*/false, /*reuse_b=*/false);
}

// ---------------------------------------------------------------------------
// Phase 1: projections.
//   Qw   = queries @ Wq^T + bq
//   Keff = keys    @ Wk^T + bk + K_time   (folds the abs-pos-K score term)
//   Veff = keys    @ Wv^T + bv + V_time   (folds the abs-pos-V output term)
// One wave per 16x16 output tile; 3 * 64 * 16 = 3072 tiles, 8 waves/block.
// ---------------------------------------------------------------------------
__global__ __launch_bounds__(256)
void proj_kernel(const float* __restrict__ queries,
                 const float* __restrict__ keys,
                 const float* __restrict__ Wq, const float* __restrict__ bq,
                 const float* __restrict__ Wk, const float* __restrict__ bk,
                 const float* __restrict__ Wv, const float* __restrict__ bv,
                 const float* __restrict__ K_time,
                 const float* __restrict__ V_time,
                 float* __restrict__ Qw,
                 float* __restrict__ Keff,
                 float* __restrict__ Veff)
{
  const int wave = threadIdx.x >> 5;
  const int lane = threadIdx.x & 31;
  const int tile = blockIdx.x * 8 + wave;
  const int nt    = tile & 15;
  const int mt    = (tile >> 4) & 63;
  const int which = tile >> 10;          // 0=Q, 1=Keff, 2=Veff

  const float* src  = (which == 0) ? queries : keys;
  const float* W    = (which == 0) ? Wq : (which == 1) ? Wk : Wv;
  const float* bias = (which == 0) ? bq : (which == 1) ? bk : bv;
  const float* add  = (which == 0) ? nullptr : (which == 1) ? K_time : V_time;
  float*       dst  = (which == 0) ? Qw : (which == 1) ? Keff : Veff;

  const int mbase = mt * 16, nbase = nt * 16;
  // 16x4 f32 A layout: lanes 0-15 hold K=0,1 ; lanes 16-31 hold K=2,3.
  const int row  = lane & 15;
  const int koff = (lane >> 4) * 2;

  v8f c = {0.f, 0.f, 0.f, 0.f, 0.f, 0.f, 0.f, 0.f};
  for (int kb = 0; kb < HH; kb += 4) {
    const float* ap = src + (size_t)(mbase + row) * HH + kb + koff;
    const float* bp = W   + (size_t)(nbase + row) * HH + kb + koff; // B[k][n] = W[n][k]
    v2f a, b;
    a[0] = ap[0]; a[1] = ap[1];
    b[0] = bp[0]; b[1] = bp[1];
    c = wmma_f32_k4(a, b, c);
  }

  // C/D layout: lanes 0-15 -> M = r, lanes 16-31 -> M = 8 + r, N = lane&15.
  const int half = lane >> 4;
  const int ncol = nbase + (lane & 15);
  const float bb = bias[ncol];
#pragma unroll
  for (int r = 0; r < 8; ++r) {
    const int m = mbase + r + half * 8;
    float v = c[r] + bb;
    if (add) v += add[(size_t)m * HH + ncol];
    dst[(size_t)m * HH + ncol] = v;
  }
}

// ---------------------------------------------------------------------------
// Phase 2: base scores (unscaled):  Sbase[b,n,q,k] = sum_d Qh[b,q,n,d]*Keffh[b,k,n,d]
// One wave per 16x16 (q,k) tile per head: 4*4*16*16 = 4096 tiles.
// ---------------------------------------------------------------------------
__global__ __launch_bounds__(256)
void score_base_kernel(const float* __restrict__ Qw,
                       const float* __restrict__ Keff,
                       float* __restrict__ Sbase)
{
  const int wave = threadIdx.x >> 5;
  const int lane = threadIdx.x & 31;
  const int tile = blockIdx.x * 8 + wave;
  const int kt = tile & 15;
  const int qt = (tile >> 4) & 15;
  const int n  = (tile >> 8) & 3;
  const int b  = tile >> 10;

  const int row  = lane & 15;
  const int koff = (lane >> 4) * 2;

  const float* qp = Qw   + ((size_t)(b * LLEN) + qt * 16 + row) * HH + n * HS;
  const float* kp = Keff + ((size_t)(b * LLEN) + kt * 16 + row) * HH + n * HS;

  v8f c = {0.f, 0.f, 0.f, 0.f, 0.f, 0.f, 0.f, 0.f};
  for (int d = 0; d < HS; d += 4) {
    v2f a, bfr;
    a[0]   = qp[d + koff]; a[1]   = qp[d + koff + 1];
    bfr[0] = kp[d + koff]; bfr[1] = kp[d + koff + 1];
    c = wmma_f32_k4(a, bfr, c);
  }

  const int half = lane >> 4;
  const int kcol = kt * 16 + (lane & 15);
#pragma unroll
  for (int r = 0; r < 8; ++r) {
    const int q = qt * 16 + r + half * 8;
    Sbase[(((size_t)(b * NHEADS + n)) * LLEN + q) * LLEN + kcol] = c[r];
  }
}

// ---------------------------------------------------------------------------
// Phase 3: fused streaming kernel. One workgroup owns 4 queries of one batch.
//   - add per-pair t-score  sum_d tK[b,q,k,:] * Q[b,q,:]   (segmented by head)
//   - mask + scale + softmax over k (rows stay in LDS)
//   - out[q,:] = sum_k A * (Veff[k,:] + tV[b,q,k,:])
// 8 wave32s: wave -> (q = wave&3, k-half = wave>>2). tK/tV read exactly once,
// with non-temporal hints so the 512MB stream doesn't evict Veff from L2.
// ---------------------------------------------------------------------------
__global__ __launch_bounds__(256)
void fused_attn_kernel(const float* __restrict__ Qw,
                       const float* __restrict__ Veff,
                       const float* __restrict__ Sbase,
                       const float* __restrict__ tK,
                       const float* __restrict__ tV,
                       const unsigned char* __restrict__ mask,
                       float* __restrict__ out)
{
  __shared__ float Qs[QT][HH];             // 4 KB
  __shared__ float Ss[NHEADS][QT][LLEN];   // 16 KB (scores -> attention probs)
  __shared__ float Outs[QT][HH];           // 4 KB

  const int b  = blockIdx.x >> 6;                // 64 blocks per batch
  const int qb = (blockIdx.x & 63) * QT;
  const int t    = threadIdx.x;
  const int wave = t >> 5;
  const int lane = t & 31;

  // stage Q tile, base scores, zero output accumulator
  for (int idx = t; idx < QT * HH; idx += 256) {
    const int q = idx >> 8, ch = idx & 255;
    Qs[q][ch]   = Qw[((size_t)(b * LLEN) + qb + q) * HH + ch];
    Outs[q][ch] = 0.f;
  }
  for (int idx = t; idx < NHEADS * QT * LLEN; idx += 256) {
    const int k = idx & 255;
    const int q = (idx >> 8) & 3;
    const int n = idx >> 10;
    Ss[n][q][k] = Sbase[(((size_t)(b * NHEADS + n)) * LLEN + (qb + q)) * LLEN + k];
  }
  __syncthreads();

  const int q   = wave & 3;
  const int kh  = wave >> 2;            // k-half: [kh*128, kh*128+128)
  const int ch0 = lane * 8;             // this lane's 8-channel chunk
  const int h   = lane >> 3;            // head of that chunk (64 ch / head)

  // ---- t-score accumulation (stream tK once, non-temporal) ----
  {
    float qreg[8];
#pragma unroll
    for (int j = 0; j < 8; ++j) qreg[j] = Qs[q][ch0 + j];

    const float* base = tK + (((size_t)(b * LLEN) + qb + q) * LLEN) * HH + ch0;
    const int k0 = kh * 128, k1 = k0 + 128;
#pragma unroll 4
    for (int k = k0; k < k1; ++k) {
      const v4f* p = (const v4f*)(base + (size_t)k * HH);
      const v4f x0 = __builtin_nontemporal_load(p);
      const v4f x1 = __builtin_nontemporal_load(p + 1);
      float s = x0[0] * qreg[0] + x0[1] * qreg[1] + x0[2] * qreg[2] + x0[3] * qreg[3]
              + x1[0] * qreg[4] + x1[1] * qreg[5] + x1[2] * qreg[6] + x1[3] * qreg[7];
      // reduce the 8 lanes covering one head segment
      s += __shfl_down(s, 4, 8);
      s += __shfl_down(s, 2, 8);
      s += __shfl_down(s, 1, 8);
      if ((lane & 7) == 0) Ss[h][q][k] += s;     // sole writer of (h,q,k)
    }
  }
  __syncthreads();

  // ---- mask + scale + softmax over k (one wave per row, 2 rows each) ----
  for (int r = wave; r < NHEADS * QT; r += 8) {
    const int n = r >> 2, qq = r & 3;
    float v[8];
#pragma unroll
    for (int j = 0; j < 8; ++j) {
      const int k = lane + j * 32;
      float s = Ss[n][qq][k] * 0.125f;           // 1/sqrt(64)
      if (mask[(size_t)(qb + qq) * LLEN + k]) s = NEG_INF;
      v[j] = s;
    }
    float m = v[0];
#pragma unroll
    for (int j = 1; j < 8; ++j) m = fmaxf(m, v[j]);
#pragma unroll
    for (int o = 16; o >= 1; o >>= 1) m = fmaxf(m, __shfl_xor(m, o, 32));
    float sum = 0.f;
#pragma unroll
    for (int j = 0; j < 8; ++j) { v[j] = __expf(v[j] - m); sum += v[j]; }
#pragma unroll
    for (int o = 16; o >= 1; o >>= 1) sum += __shfl_xor(sum, o, 32);
    const float inv = 1.0f / sum;
#pragma unroll
    for (int j = 0; j < 8; ++j) Ss[n][qq][lane + j * 32] = v[j] * inv;
  }
  __syncthreads();

  // ---- output accumulation (stream tV once, non-temporal; Veff rides in L2) ----
  {
    float acc[8] = {0.f, 0.f, 0.f, 0.f, 0.f, 0.f, 0.f, 0.f};
    const float* tvb = tV + (((size_t)(b * LLEN) + qb + q) * LLEN) * HH + ch0;
    const float* veb = Veff + ((size_t)(b * LLEN)) * HH + ch0;
    const int k0 = kh * 128, k1 = k0 + 128;
#pragma unroll 4
    for (int k = k0; k < k1; ++k) {
      const float a = Ss[h][q][k];
      const v4f* pv = (const v4f*)(tvb + (size_t)k * HH);
      const v4f* pe = (const v4f*)(veb + (size_t)k * HH);
      const v4f t0 = __builtin_nontemporal_load(pv);
      const v4f t1 = __builtin_nontemporal_load(pv + 1);
      const v4f e0 = pe[0], e1 = pe[1];
      acc[0] += a * (t0[0] + e0[0]); acc[1] += a * (t0[1] + e0[1]);
      acc[2] += a * (t0[2] + e0[2]); acc[3] += a * (t0[3] + e0[3]);
      acc[4] += a * (t1[0] + e1[0]); acc[5] += a * (t1[1] + e1[1]);
      acc[6] += a * (t1[2] + e1[2]); acc[7] += a * (t1[3] + e1[3]);
    }
    // merge the two k-halves via LDS float atomics (ds_add_f32)
#pragma unroll
    for (int j = 0; j < 8; ++j) atomicAdd(&Outs[q][ch0 + j], acc[j]);
  }
  __syncthreads();

  // final result is written once and never re-read -> non-temporal store
  for (int idx = t; idx < QT * HH / 4; idx += 256) {
    const int qq = (idx * 4) >> 8, ch = (idx * 4) & 255;
    v4f o;
    o[0] = Outs[qq][ch + 0]; o[1] = Outs[qq][ch + 1];
    o[2] = Outs[qq][ch + 2]; o[3] = Outs[qq][ch + 3];
    __builtin_nontemporal_store(
        o, (v4f*)(out + ((size_t)(b * LLEN) + qb + qq) * HH + ch));
  }
}

// ---------------------------------------------------------------------------
// Launch. Workspace layout (floats):
//   Qw    [0,       262144)   1 MB
//   Keff  [262144,  524288)   1 MB
//   Veff  [524288,  786432)   1 MB
//   Sbase [786432, 1835008)   4 MB   (B*HEADS*L*L)
// ---------------------------------------------------------------------------
extern "C" void kernel_launch(void* const* d_in, const int* in_sizes, int n_in,
                              void* d_out, int out_size, void* d_ws, size_t ws_size,
                              hipStream_t stream)
{
  const float*         queries = (const float*)d_in[0];
  const float*         keys    = (const float*)d_in[1];
  const unsigned char* mask    = (const unsigned char*)d_in[2];
  const float*         tK      = (const float*)d_in[3];
  const float*         tV      = (const float*)d_in[4];
  const float*         K_time  = (const float*)d_in[5];
  const float*         V_time  = (const float*)d_in[6];
  const float*         Wq      = (const float*)d_in[7];
  const float*         bq      = (const float*)d_in[8];
  const float*         Wk      = (const float*)d_in[9];
  const float*         bk      = (const float*)d_in[10];
  const float*         Wv      = (const float*)d_in[11];
  const float*         bv      = (const float*)d_in[12];
  float* out = (float*)d_out;

  float* ws    = (float*)d_ws;
  float* Qw    = ws;
  float* Keff  = ws + 262144;
  float* Veff  = ws + 524288;
  float* Sbase = ws + 786432;

  proj_kernel<<<384, 256, 0, stream>>>(queries, keys, Wq, bq, Wk, bk, Wv, bv,
                                       K_time, V_time, Qw, Keff, Veff);
  score_base_kernel<<<512, 256, 0, stream>>>(Qw, Keff, Sbase);
  fused_attn_kernel<<<BB * (LLEN / QT), 256, 0, stream>>>(Qw, Veff, Sbase,
                                                          tK, tV, mask, out);
}